// MetricLoss_68315749810783
// MI455X (gfx1250) — compile-verified
//
#include <hip/hip_runtime.h>
#include <math.h>

// CDNA5 / gfx1250: wave32, WMMA f32 16x16x4 path.
typedef float v2f __attribute__((ext_vector_type(2)));
typedef float v8f __attribute__((ext_vector_type(8)));

#define N_PTS  4096
#define FEAT   512
#define MARGIN 1.0f
#define LDSPITCH 516   // 512 + 4 pad floats: conflict-free ds_load_b64 fragments

// Workspace layout (float units)
#define WS_RNS   0        // 4096 floats: row_negsum
#define WS_SQA   4096     // 4096 floats: |a_j|^2
#define WS_SQB   8192     // 4096 floats: |b_i|^2
#define WS_ACC   12288    // 1 float: loss accumulator
#define WS_HIST  12296    // 16 ints: label histogram
#define WS_ZERO_COUNT 12320

// ---------------------------------------------------------------- utilities
__global__ void ml_zero_ws(float* ws) {
  int t = blockIdx.x * blockDim.x + threadIdx.x;
  if (t < WS_ZERO_COUNT) ws[t] = 0.0f;
}

// One wave32 per row; rows [0,4096) -> a, [4096,8192) -> b.
__global__ void ml_norms(const float* __restrict__ a, const float* __restrict__ b,
                         float* __restrict__ ws) {
  int wave = (blockIdx.x * blockDim.x + threadIdx.x) >> 5;
  int lane = threadIdx.x & 31;
  const float* src;
  float* dst;
  int row;
  if (wave < N_PTS) { src = a; dst = ws + WS_SQA; row = wave; }
  else              { src = b; dst = ws + WS_SQB; row = wave - N_PTS; }
  const float* p = src + (size_t)row * FEAT;
  float s = 0.0f;
  for (int k = lane; k < FEAT; k += 32) { float v = p[k]; s += v * v; }
  #pragma unroll
  for (int m = 16; m >= 1; m >>= 1) s += __shfl_xor(s, m, 32);
  if (lane == 0) dst[row] = s;
}

__global__ void ml_hist(const int* __restrict__ labels, float* __restrict__ ws) {
  int t = blockIdx.x * blockDim.x + threadIdx.x;
  int* hist = (int*)(ws + WS_HIST);
  if (t < N_PTS) atomicAdd(&hist[labels[t]], 1);
}

// Stage the block's 16-row b slab into padded LDS (one time per block).
__device__ __forceinline__ void stage_b_slab(float* ldsb, const float* __restrict__ b,
                                             int i0) {
  for (int idx = threadIdx.x; idx < 16 * (FEAT / 4); idx += 256) {
    int row  = idx >> 7;        // FEAT/4 = 128 float4 per row
    int col4 = idx & 127;
    float4 v = *(const float4*)(b + (size_t)(i0 + row) * FEAT + col4 * 4);
    *(float4*)(&ldsb[row * LDSPITCH + col4 * 4]) = v;
  }
  __syncthreads();
}

// Software-pipelined 4-tile GEMM chunk: A fragment from LDS (shared slab),
// 4 independent B fragments from global, 4 independent WMMA chains.
// Unroll kept moderate: the backend builds its own deep cross-iteration
// pipeline; deeper manual unroll caused scratch spills in the hot loop.
__device__ __forceinline__ void gemm_tile4(const float* __restrict__ ldsA,
                                           const float* __restrict__ pB0,
                                           const float* __restrict__ pB1,
                                           const float* __restrict__ pB2,
                                           const float* __restrict__ pB3,
                                           v8f c[4]) {
  v2f aC = *(const v2f*)(ldsA);
  v2f b0 = *(const v2f*)(pB0);
  v2f b1 = *(const v2f*)(pB1);
  v2f b2 = *(const v2f*)(pB2);
  v2f b3 = *(const v2f*)(pB3);
  #pragma unroll 4
  for (int k = 4; k < FEAT; k += 4) {
    // issue next-iteration fragment loads before consuming current ones
    v2f aN = *(const v2f*)(ldsA + k);
    v2f n0 = *(const v2f*)(pB0 + k);
    v2f n1 = *(const v2f*)(pB1 + k);
    v2f n2 = *(const v2f*)(pB2 + k);
    v2f n3 = *(const v2f*)(pB3 + k);
    c[0] = __builtin_amdgcn_wmma_f32_16x16x4_f32(false, aC, false, b0, (short)0, c[0], false, false);
    c[1] = __builtin_amdgcn_wmma_f32_16x16x4_f32(false, aC, false, b1, (short)0, c[1], false, false);
    c[2] = __builtin_amdgcn_wmma_f32_16x16x4_f32(false, aC, false, b2, (short)0, c[2], false, false);
    c[3] = __builtin_amdgcn_wmma_f32_16x16x4_f32(false, aC, false, b3, (short)0, c[3], false, false);
    aC = aN; b0 = n0; b1 = n1; b2 = n2; b3 = n3;
  }
  c[0] = __builtin_amdgcn_wmma_f32_16x16x4_f32(false, aC, false, b0, (short)0, c[0], false, false);
  c[1] = __builtin_amdgcn_wmma_f32_16x16x4_f32(false, aC, false, b1, (short)0, c[1], false, false);
  c[2] = __builtin_amdgcn_wmma_f32_16x16x4_f32(false, aC, false, b2, (short)0, c[2], false, false);
  c[3] = __builtin_amdgcn_wmma_f32_16x16x4_f32(false, aC, false, b3, (short)0, c[3], false, false);
}

// ------------------------------------------------- pass 1: masked row sums
// Block owns a 16-row slab of D (rows of b) staged in LDS; its 8 waves cover
// the 256 column tiles in groups of 4 concurrent tiles.
__global__ void __launch_bounds__(256)
ml_negsum(const float* __restrict__ a, const float* __restrict__ b,
          const int* __restrict__ labels, float* __restrict__ ws) {
  __shared__ float ldsb[16 * LDSPITCH];
  const int i0     = blockIdx.x * 16;
  const int waveId = threadIdx.x >> 5;
  const int lane   = threadIdx.x & 31;
  const int half   = lane >> 4;
  const int lrow   = lane & 15;

  const float* __restrict__ sqa = ws + WS_SQA;
  const float* __restrict__ sqb = ws + WS_SQB;
  float* __restrict__ rns = ws + WS_RNS;

  stage_b_slab(ldsb, b, i0);
  const float* ldsA = &ldsb[lrow * LDSPITCH + half * 2];

  int   labi[8];
  float bbv[8];
  #pragma unroll
  for (int r = 0; r < 8; ++r) {
    int i = i0 + r + 8 * half;
    labi[r] = labels[i];
    bbv[r]  = sqb[i];
  }

  float rowacc[8];
  #pragma unroll
  for (int r = 0; r < 8; ++r) rowacc[r] = 0.0f;

  for (int g = 0; g < 8; ++g) {               // 8 groups of 4 tiles per wave
    const int jt0 = waveId * 4 + g * 32;      // tiles jt0..jt0+3
    const float* pB[4];
    #pragma unroll
    for (int t = 0; t < 4; ++t)
      pB[t] = a + (size_t)((jt0 + t) * 16 + lrow) * FEAT + half * 2;

    v8f c[4] = {v8f{}, v8f{}, v8f{}, v8f{}};
    gemm_tile4(ldsA, pB[0], pB[1], pB[2], pB[3], c);

    #pragma unroll
    for (int t = 0; t < 4; ++t) {
      const int   j    = (jt0 + t) * 16 + lrow;
      const int   labj = labels[j];
      const float aav  = sqa[j];
      #pragma unroll
      for (int r = 0; r < 8; ++r) {
        float dsq = bbv[r] + aav - 2.0f * c[t][r];
        float D   = sqrtf(fmaxf(dsq, 0.0f));
        float v   = (labj != labi[r]) ? expf(MARGIN - D) : 0.0f;
        rowacc[r] += v;
      }
    }
  }

  // Reduce each row over the 16 lanes of its half-wave, then one atomic/row.
  #pragma unroll
  for (int r = 0; r < 8; ++r) {
    float s = rowacc[r];
    s += __shfl_xor(s, 1, 32);
    s += __shfl_xor(s, 2, 32);
    s += __shfl_xor(s, 4, 32);
    s += __shfl_xor(s, 8, 32);
    if (lrow == 0) atomicAdd(&rns[i0 + r + 8 * half], s);
  }
}

// ------------------------------------------------- pass 2: hinge-squared sum
__global__ void __launch_bounds__(256)
ml_loss(const float* __restrict__ a, const float* __restrict__ b,
        const int* __restrict__ labels, float* __restrict__ ws) {
  __shared__ float ldsb[16 * LDSPITCH];
  const int i0     = blockIdx.x * 16;
  const int waveId = threadIdx.x >> 5;
  const int lane   = threadIdx.x & 31;
  const int half   = lane >> 4;
  const int lrow   = lane & 15;

  const float* __restrict__ sqa = ws + WS_SQA;
  const float* __restrict__ sqb = ws + WS_SQB;
  const float* __restrict__ rns = ws + WS_RNS;

  stage_b_slab(ldsb, b, i0);
  const float* ldsA = &ldsb[lrow * LDSPITCH + half * 2];

  int   labi[8];
  float bbv[8];
  float rnsi[8];
  #pragma unroll
  for (int r = 0; r < 8; ++r) {
    int i = i0 + r + 8 * half;
    labi[r] = labels[i];
    bbv[r]  = sqb[i];
    rnsi[r] = rns[i];
  }

  float ls = 0.0f;

  for (int g = 0; g < 8; ++g) {
    const int jt0 = waveId * 4 + g * 32;
    const float* pB[4];
    #pragma unroll
    for (int t = 0; t < 4; ++t)
      pB[t] = a + (size_t)((jt0 + t) * 16 + lrow) * FEAT + half * 2;

    v8f c[4] = {v8f{}, v8f{}, v8f{}, v8f{}};
    gemm_tile4(ldsA, pB[0], pB[1], pB[2], pB[3], c);

    #pragma unroll
    for (int t = 0; t < 4; ++t) {
      const int   j    = (jt0 + t) * 16 + lrow;
      const int   labj = labels[j];
      const float aav  = sqa[j];
      const float rnsj = rns[j];
      #pragma unroll
      for (int r = 0; r < 8; ++r) {
        const int i = i0 + r + 8 * half;
        float dsq = bbv[r] + aav - 2.0f * c[t][r];
        float D   = sqrtf(fmaxf(dsq, 0.0f));
        float v   = logf(rnsi[r] + rnsj) + D;
        float h   = fmaxf(v, 0.0f);
        float contrib = h * h;
        bool  pos = (labj == labi[r]) && (i != j);
        ls += pos ? contrib : 0.0f;
      }
    }
  }

  // Full-wave reduction, one atomic per wave.
  #pragma unroll
  for (int m = 16; m >= 1; m >>= 1) ls += __shfl_xor(ls, m, 32);
  if (lane == 0) atomicAdd(ws + WS_ACC, ls);
}

__global__ void ml_finalize(const float* __restrict__ ws, float* __restrict__ out) {
  const int* hist = (const int*)(ws + WS_HIST);
  long long np = 0;
  #pragma unroll
  for (int l = 0; l < 16; ++l) {
    long long c = hist[l];
    np += c * (c - 1);  // ordered positive pairs per class
  }
  out[0] = ws[WS_ACC] / (2.0f * (float)np);
}

// ------------------------------------------------------------------ launch
extern "C" void kernel_launch(void* const* d_in, const int* in_sizes, int n_in,
                              void* d_out, int out_size, void* d_ws, size_t ws_size,
                              hipStream_t stream) {
  (void)in_sizes; (void)n_in; (void)out_size; (void)ws_size;
  const float* a      = (const float*)d_in[0];
  const float* b      = (const float*)d_in[1];
  const int*   labels = (const int*)d_in[2];
  float*       out    = (float*)d_out;
  float*       ws     = (float*)d_ws;

  ml_zero_ws<<<(WS_ZERO_COUNT + 255) / 256, 256, 0, stream>>>(ws);
  ml_norms<<<(2 * N_PTS) / 8, 256, 0, stream>>>(a, b, ws);     // 8 wave32/block
  ml_hist<<<N_PTS / 256, 256, 0, stream>>>(labels, ws);
  ml_negsum<<<N_PTS / 16, 256, 0, stream>>>(a, b, labels, ws);
  ml_loss<<<N_PTS / 16, 256, 0, stream>>>(a, b, labels, ws);
  ml_finalize<<<1, 1, 0, stream>>>(ws, out);
}